// MultiHeadAttention_37718402794069
// MI455X (gfx1250) — compile-verified
//
#include <hip/hip_runtime.h>

typedef float v2f __attribute__((ext_vector_type(2)));
typedef float v8f __attribute__((ext_vector_type(8)));

#define B_  8
#define S_  2048
#define D_  128
#define H_  4
#define DK_ 32

static __device__ __forceinline__ v8f wmma4(v2f a, v2f b, v8f c) {
  // V_WMMA_F32_16X16X4_F32: D = A(16x4) x B(4x16) + C(16x16), f32 throughout
  return __builtin_amdgcn_wmma_f32_16x16x4_f32(false, a, false, b, (short)0, c,
                                               false, false);
}

// ---------------------------------------------------------------------------
// Kernel 1: the four input projections.
//   which = 0: Resid = inQ @ W0    1: Qp = inQ @ Wq
//           2: Kp    = inK @ Wk    3: Vp = inV @ Wv
// Each block: one 16-row tile x all 128 cols; 8 waves, one 16-col tile each.
// ---------------------------------------------------------------------------
__global__ __launch_bounds__(256) void proj_kernel(
    const float* __restrict__ inQ, const float* __restrict__ inK,
    const float* __restrict__ inV, const float* __restrict__ W0,
    const float* __restrict__ Wq, const float* __restrict__ Wk,
    const float* __restrict__ Wv, float* __restrict__ Qp,
    float* __restrict__ Kp, float* __restrict__ Vp,
    float* __restrict__ Resid) {
  const int lane = threadIdx.x & 31;
  const int wave = threadIdx.x >> 5;
  const int row0 = blockIdx.x * 16;
  const int which = blockIdx.y;

  const float* X; const float* W; float* Y;
  if (which == 0)      { X = inQ; W = W0; Y = Resid; }
  else if (which == 1) { X = inQ; W = Wq; Y = Qp; }
  else if (which == 2) { X = inK; W = Wk; Y = Kp; }
  else                 { X = inV; W = Wv; Y = Vp; }

  const int m  = lane & 15;
  const int kl = (lane >> 4) << 1;   // 0 or 2
  const int mg = (lane >> 4) << 3;   // 0 or 8
  const int col = wave * 16 + m;

  const float* xrow = X + (size_t)(row0 + m) * D_;
  v8f acc = {};
#pragma unroll 4
  for (int kk = 0; kk < D_; kk += 4) {
    v2f a = *(const v2f*)(xrow + kk + kl);
    v2f b;
    b.x = W[(size_t)(kk + kl) * D_ + col];
    b.y = W[(size_t)(kk + kl + 1) * D_ + col];
    acc = wmma4(a, b, acc);
  }
#pragma unroll
  for (int i = 0; i < 8; ++i)
    Y[(size_t)(row0 + mg + i) * D_ + col] = acc[i];
}

// ---------------------------------------------------------------------------
// Kernel 2: fused attention core for one (b, h, 16-query tile).
//   scores = Q K^T / sqrt(dk), mask->-1e9, softmax, write attn (coalesced),
//   context = P V (unnormalized P, scaled by 1/rowsum at the end).
// 16x2048 score strip lives in LDS (padded stride to kill bank conflicts).
// ---------------------------------------------------------------------------
#define STRIP_ 2052  // 2048 + 4 pad floats per row

__global__ __launch_bounds__(256) void attn_kernel(
    const float* __restrict__ Qp, const float* __restrict__ Kp,
    const float* __restrict__ Vp, const unsigned char* __restrict__ mask,
    float* __restrict__ attn_out, float* __restrict__ Ctx) {
  __shared__ float strip[16 * STRIP_];     // 131 KB: exp(scores - rowmax)
  __shared__ float red[256];
  __shared__ float rowmax[16];
  __shared__ float rowinv[16];
  __shared__ float ctxp[8][16][32];        // per-wave partial contexts

  const int lane = threadIdx.x & 31;
  const int wave = threadIdx.x >> 5;
  const int qt = blockIdx.x;               // 0..127
  const int h  = blockIdx.y;
  const int b  = blockIdx.z;
  const int q0 = qt * 16;

  const int m  = lane & 15;
  const int kl = (lane >> 4) << 1;         // 0 or 2
  const int mg = (lane >> 4) << 3;         // 0 or 8

  // Preload the 16x32 Q tile in WMMA-A layout (8 k-steps of 4).
  const float* qbase = Qp + ((size_t)(b * S_ + q0 + m) * D_ + h * DK_);
  v2f aq[8];
#pragma unroll
  for (int d = 0; d < 8; ++d)
    aq[d] = *(const v2f*)(qbase + 4 * d + kl);

  const float scale = 0.17677669529663687f;  // 1/sqrt(32)
  const size_t headbase = (size_t)b * S_ * D_ + (size_t)h * DK_;
  const unsigned char* mbase = mask + (size_t)b * S_ * S_;

  // ---- scores: wave w handles key tiles j = w, w+8, ... (16 tiles each) ----
  for (int j = wave; j < S_ / 16; j += 8) {
    v8f acc = {};
    const float* krow = Kp + headbase + (size_t)(j * 16 + m) * D_;  // key n=m
#pragma unroll
    for (int d = 0; d < 8; ++d) {
      v2f bb = *(const v2f*)(krow + 4 * d + kl);
      acc = wmma4(aq[d], bb, acc);
    }
#pragma unroll
    for (int i = 0; i < 8; ++i) {
      const int qm = mg + i;
      const int kc = j * 16 + m;
      float s = acc[i] * scale;
      if (mbase[(size_t)(q0 + qm) * S_ + kc]) s = -1e9f;
      strip[qm * STRIP_ + kc] = s;
    }
  }
  __syncthreads();

  // ---- softmax: row r handled by 16 threads (thread = r*16 + c0) ----
  const int r  = threadIdx.x >> 4;
  const int c0 = threadIdx.x & 15;
  float pmax = -3.4e38f;
  for (int c = c0; c < S_; c += 16)
    pmax = fmaxf(pmax, strip[r * STRIP_ + c]);
  red[threadIdx.x] = pmax;
  __syncthreads();
  if (c0 == 0) {
    float mx = red[r * 16];
#pragma unroll
    for (int i = 1; i < 16; ++i) mx = fmaxf(mx, red[r * 16 + i]);
    rowmax[r] = mx;
  }
  __syncthreads();
  const float mx = rowmax[r];
  float psum = 0.f;
  for (int c = c0; c < S_; c += 16) {
    float v = __expf(strip[r * STRIP_ + c] - mx);
    strip[r * STRIP_ + c] = v;
    psum += v;
  }
  red[threadIdx.x] = psum;
  __syncthreads();
  if (c0 == 0) {
    float sm = 0.f;
#pragma unroll
    for (int i = 0; i < 16; ++i) sm += red[r * 16 + i];
    rowinv[r] = 1.0f / sm;
  }
  __syncthreads();

  // ---- coalesced attention write (normalization folded in) ----
  {
    float* ab = attn_out + ((size_t)(b * H_ + h) * S_ + q0) * S_;
    for (int idx = threadIdx.x; idx < 16 * S_; idx += 256) {
      const int rr = idx >> 11;            // /2048
      const int cc = idx & (S_ - 1);
      ab[idx] = strip[rr * STRIP_ + cc] * rowinv[rr];
    }
  }

  // ---- context = P(16x2048) @ V(2048x32); wave w covers k in [256w,256w+256)
  v8f ca = {}, cb = {};
  const float* vb = Vp + headbase;
  const int kbeg = wave * 256;
#pragma unroll 4
  for (int k0 = kbeg; k0 < kbeg + 256; k0 += 4) {
    v2f a = *(const v2f*)&strip[m * STRIP_ + k0 + kl];
    const float* v0 = vb + (size_t)(k0 + kl) * D_;
    v2f b0, b1;
    b0.x = v0[m];       b0.y = v0[D_ + m];
    b1.x = v0[16 + m];  b1.y = v0[D_ + 16 + m];
    ca = wmma4(a, b0, ca);
    cb = wmma4(a, b1, cb);
  }
#pragma unroll
  for (int i = 0; i < 8; ++i) {
    ctxp[wave][mg + i][m]      = ca[i];
    ctxp[wave][mg + i][16 + m] = cb[i];
  }
  __syncthreads();

  // cross-wave reduce, scale by 1/rowsum, store Ctx[b, q, h*32+n]
  for (int e = threadIdx.x; e < 16 * 32; e += 256) {
    const int qm = e >> 5;
    const int n  = e & 31;
    float s = 0.f;
#pragma unroll
    for (int w = 0; w < 8; ++w) s += ctxp[w][qm][n];
    Ctx[(size_t)(b * S_ + q0 + qm) * D_ + h * DK_ + n] = s * rowinv[qm];
  }
}

// ---------------------------------------------------------------------------
// Kernel 3: Out = LayerNorm(Ctx @ Wo + Resid)
// ---------------------------------------------------------------------------
__global__ __launch_bounds__(256) void out_ln_kernel(
    const float* __restrict__ Ctx, const float* __restrict__ Wo,
    const float* __restrict__ Resid, float* __restrict__ Out) {
  __shared__ float y[16 * 132];
  __shared__ float red1[256];
  __shared__ float red2[256];
  __shared__ float mu[16];
  __shared__ float inv[16];

  const int lane = threadIdx.x & 31;
  const int wave = threadIdx.x >> 5;
  const int row0 = blockIdx.x * 16;
  const int m  = lane & 15;
  const int kl = (lane >> 4) << 1;
  const int mg = (lane >> 4) << 3;
  const int col = wave * 16 + m;

  const float* xrow = Ctx + (size_t)(row0 + m) * D_;
  v8f acc = {};
#pragma unroll 4
  for (int kk = 0; kk < D_; kk += 4) {
    v2f a = *(const v2f*)(xrow + kk + kl);
    v2f b;
    b.x = Wo[(size_t)(kk + kl) * D_ + col];
    b.y = Wo[(size_t)(kk + kl + 1) * D_ + col];
    acc = wmma4(a, b, acc);
  }
#pragma unroll
  for (int i = 0; i < 8; ++i) {
    const int rr = mg + i;
    y[rr * 132 + col] = acc[i] + Resid[(size_t)(row0 + rr) * D_ + col];
  }
  __syncthreads();

  const int r  = threadIdx.x >> 4;
  const int c0 = threadIdx.x & 15;
  float s = 0.f, s2 = 0.f;
  for (int c = c0; c < D_; c += 16) {
    const float v = y[r * 132 + c];
    s += v; s2 += v * v;
  }
  red1[threadIdx.x] = s;
  red2[threadIdx.x] = s2;
  __syncthreads();
  if (c0 == 0) {
    float ts = 0.f, ts2 = 0.f;
#pragma unroll
    for (int i = 0; i < 16; ++i) { ts += red1[r * 16 + i]; ts2 += red2[r * 16 + i]; }
    const float mean = ts * (1.0f / D_);
    const float var  = ts2 * (1.0f / D_) - mean * mean;
    mu[r]  = mean;
    inv[r] = rsqrtf(var + 1e-5f);
  }
  __syncthreads();
  for (int idx = threadIdx.x; idx < 16 * D_; idx += 256) {
    const int rr = idx >> 7;
    const int cc = idx & (D_ - 1);
    Out[(size_t)row0 * D_ + idx] = (y[rr * 132 + cc] - mu[rr]) * inv[rr];
  }
}

// ---------------------------------------------------------------------------
extern "C" void kernel_launch(void* const* d_in, const int* in_sizes, int n_in,
                              void* d_out, int out_size, void* d_ws,
                              size_t ws_size, hipStream_t stream) {
  const float* inQ = (const float*)d_in[0];
  const float* inK = (const float*)d_in[1];
  const float* inV = (const float*)d_in[2];
  const unsigned char* mask = (const unsigned char*)d_in[3];  // jax bool = 1B
  const float* W0 = (const float*)d_in[4];
  const float* Wq = (const float*)d_in[5];
  const float* Wk = (const float*)d_in[6];
  const float* Wv = (const float*)d_in[7];
  const float* Wo = (const float*)d_in[8];

  const size_t NE = (size_t)B_ * S_ * D_;  // 2,097,152 floats
  float* ws = (float*)d_ws;
  float* Qp    = ws;
  float* Kp    = ws + NE;
  float* Vp    = ws + 2 * NE;
  float* Resid = ws + 3 * NE;
  float* Ctx   = ws + 4 * NE;              // 40 MB total scratch

  float* out_ln = (float*)d_out;           // [B,S,D]
  float* attn   = out_ln + NE;             // [B,H,S,S]

  proj_kernel<<<dim3(1024, 4), 256, 0, stream>>>(inQ, inK, inV, W0, Wq, Wk, Wv,
                                                 Qp, Kp, Vp, Resid);
  attn_kernel<<<dim3(128, 4, 8), 256, 0, stream>>>(Qp, Kp, Vp, mask, attn, Ctx);
  out_ln_kernel<<<1024, 256, 0, stream>>>(Ctx, Wo, Resid, out_ln);
}